// NAE_39642548142693
// MI455X (gfx1250) — compile-verified
//
#include <hip/hip_runtime.h>
#include <hip/hip_bf16.h>

// NAE reduction: sum(|pred - gt| / gt) / n_sample over 16384x2048 f32.
// Bandwidth-bound (256 MiB in, 1 float out; ~11.5 us floor at 23.3 TB/s).
// Pass 1: grid-stride b128 NT loads, per-lane terms fed into
//         V_WMMA_F32_16X16X4_F32 with B=ones as an XDL-side accumulator.
//         Hot loop is UNGUARDED (uniform full trip count for the whole grid,
//         EXEC all-ones); a single guarded epilogue iteration covers the
//         ragged remainder with ones-substitution so the WMMA still sees
//         full EXEC. Wave butterfly + LDS reduce -> one partial per block.
// Pass 2: single block deterministically reduces partials -> d_out[0].

typedef float v2f   __attribute__((ext_vector_type(2)));
typedef float v8f   __attribute__((ext_vector_type(8)));
typedef float f32x4 __attribute__((ext_vector_type(4)));

static __device__ __forceinline__ float nae_term(float p, float g) {
    // |p - g| / g via v_rcp_f32 (1 ulp); divide precision is irrelevant next
    // to the f32 accumulation noise of a 33M-element sum.
    return __builtin_fabsf(p - g) * __builtin_amdgcn_rcpf(g);
}

__global__ void __launch_bounds__(256)
nae_partial_kernel(const float* __restrict__ pred,
                   const float* __restrict__ gt,
                   float* __restrict__ partials,
                   long long n4) {
    const long long tid    = (long long)blockIdx.x * blockDim.x + threadIdx.x;
    const long long stride = (long long)gridDim.x * blockDim.x;
    const long long full   = n4 / stride;   // unguarded iterations (uniform)

    const f32x4* __restrict__ p4 = (const f32x4*)pred;
    const f32x4* __restrict__ g4 = (const f32x4*)gt;

    v8f c = {0.f, 0.f, 0.f, 0.f, 0.f, 0.f, 0.f, 0.f};
    v2f bones; bones[0] = 1.0f; bones[1] = 1.0f;   // B = ones(4x16)

    long long i = tid;
    for (long long it = 0; it < full; ++it, i += stride) {
        const f32x4 p = __builtin_nontemporal_load(&p4[i]); // b128 TH_LOAD_NT
        const f32x4 g = __builtin_nontemporal_load(&g4[i]);
        v2f a0; a0[0] = nae_term(p.x, g.x); a0[1] = nae_term(p.y, g.y);
        v2f a1; a1[0] = nae_term(p.z, g.z); a1[1] = nae_term(p.w, g.w);
        // D[m][n] = C[m][n] + sum_k A[m][k]: accumulates 64 lane-values per
        // instruction; all 16 N-columns of D stay bitwise identical.
        c = __builtin_amdgcn_wmma_f32_16x16x4_f32(
                false, a0, false, bones, (short)0, c, false, false);
        c = __builtin_amdgcn_wmma_f32_16x16x4_f32(
                false, a1, false, bones, (short)0, c, false, false);
    }

    // Single guarded epilogue iteration for n4 % stride lanes (exactly zero
    // for the 16384x2048 case). All threads execute it; out-of-range lanes
    // contribute |1-1|/1 = 0, keeping EXEC full at the WMMA.
    {
        f32x4 p = {1.f, 1.f, 1.f, 1.f};
        f32x4 g = {1.f, 1.f, 1.f, 1.f};
        if (i < n4) {
            p = __builtin_nontemporal_load(&p4[i]);
            g = __builtin_nontemporal_load(&g4[i]);
        }
        v2f a0; a0[0] = nae_term(p.x, g.x); a0[1] = nae_term(p.y, g.y);
        v2f a1; a1[0] = nae_term(p.z, g.z); a1[1] = nae_term(p.w, g.w);
        c = __builtin_amdgcn_wmma_f32_16x16x4_f32(
                false, a0, false, bones, (short)0, c, false, false);
        c = __builtin_amdgcn_wmma_f32_16x16x4_f32(
                false, a1, false, bones, (short)0, c, false, false);
    }

    // Sum this lane's 8 D registers, butterfly-reduce the wave, then divide by
    // 16 (exact) to undo the 16x column replication.
    float s = c[0] + c[1] + c[2] + c[3] + c[4] + c[5] + c[6] + c[7];
#pragma unroll
    for (int off = 16; off > 0; off >>= 1)
        s += __shfl_xor(s, off, 32);
    s *= 0.0625f;

    __shared__ float wsum[8];
    const int lane = threadIdx.x & 31;
    const int wave = threadIdx.x >> 5;
    if (lane == 0) wsum[wave] = s;
    __syncthreads();
    if (threadIdx.x == 0) {
        float b = 0.f;
#pragma unroll
        for (int w = 0; w < 8; ++w) b += wsum[w];
        partials[blockIdx.x] = b;
    }
}

__global__ void __launch_bounds__(256)
nae_final_kernel(const float* __restrict__ partials, int nPartials,
                 const float* __restrict__ pred, const float* __restrict__ gt,
                 long long n4, long long n, float inv_nsample,
                 float* __restrict__ out) {
    float s = 0.f;
    for (int i = threadIdx.x; i < nPartials; i += 256)
        s += partials[i];
#pragma unroll
    for (int off = 16; off > 0; off >>= 1)
        s += __shfl_xor(s, off, 32);

    __shared__ float wsum[8];
    const int lane = threadIdx.x & 31;
    const int wave = threadIdx.x >> 5;
    if (lane == 0) wsum[wave] = s;
    __syncthreads();
    if (threadIdx.x == 0) {
        float t = 0.f;
#pragma unroll
        for (int w = 0; w < 8; ++w) t += wsum[w];
        // Scalar tail for n not divisible by 4 (none for 16384x2048).
        for (long long i = n4 * 4; i < n; ++i)
            t += nae_term(pred[i], gt[i]);
        out[0] = t * inv_nsample;
    }
}

extern "C" void kernel_launch(void* const* d_in, const int* in_sizes, int n_in,
                              void* d_out, int out_size, void* d_ws, size_t ws_size,
                              hipStream_t stream) {
    const float* pred = (const float*)d_in[0];
    const float* gt   = (const float*)d_in[1];
    const long long n  = (long long)in_sizes[0];   // 16384*2048 = 33,554,432
    const long long n4 = n >> 2;

    float* partials = (float*)d_ws;
    int blocks = 1024;                              // 262,144 threads, 32 iters each
    if ((size_t)blocks * sizeof(float) > ws_size)
        blocks = (int)(ws_size / sizeof(float));
    if (blocks < 1) blocks = 1;

    nae_partial_kernel<<<blocks, 256, 0, stream>>>(pred, gt, partials, n4);

    // n_sample = reference shape[0] = 16384 (not recoverable from flat sizes).
    const float inv_nsample = 1.0f / 16384.0f;
    nae_final_kernel<<<1, 256, 0, stream>>>(partials, blocks, pred, gt,
                                            n4, n, inv_nsample, (float*)d_out);
}